// SoftRank_6425271075476
// MI455X (gfx1250) — compile-verified
//
#include <hip/hip_runtime.h>

// ---------------------------------------------------------------------------
// SoftRank: out[b,j,c] = (1/N) * sum_i sigmoid(ALPHA * (x[b,j,c] - x[b,i,c]))
// B=8, N=1024, C=16, fp32.  Transcendental-bound (134M sigmoids); I/O ~1MB is
// negligible at 23.3 TB/s.  One batch (64KB) is async-copied to LDS
// (global_load_async_to_lds_b128), then each thread reduces over all i for
// one j-row / 2 channels at 3 ops per sigmoid via gfx1250 v_tanh_f32:
//   sigmoid(2z) = 0.5 + 0.5*tanh(z),  z = (ALPHA/2)*(xj - xi) = fma(xi,-a2,cj)
// ---------------------------------------------------------------------------

namespace {
constexpr int   kB        = 8;
constexpr int   kN        = 1024;
constexpr int   kC        = 16;
constexpr int   kThreads  = 256;
constexpr int   kCPT      = 2;                   // channels per thread
constexpr int   kCG       = kC / kCPT;           // 8 channel groups
constexpr int   kJPerBlk  = kThreads / kCG;      // 32 j rows per block
constexpr int   kJBlks    = kN / kJPerBlk;       // 32 j-blocks per batch
constexpr float kAlphaH   = 500.0f;              // ALPHA / 2
}

// ---- CDNA5 async global->LDS copy (16B per lane), with asm fallback --------
typedef int v4i __attribute__((ext_vector_type(4)));
typedef __attribute__((address_space(1))) v4i* as1_v4i_ptr;
typedef __attribute__((address_space(3))) v4i* as3_v4i_ptr;

__device__ __forceinline__ void async_cp16(void* lds_dst, const void* gsrc) {
#if __has_builtin(__builtin_amdgcn_global_load_async_to_lds_b128)
  __builtin_amdgcn_global_load_async_to_lds_b128(
      (as1_v4i_ptr)(void*)(gsrc),
      (as3_v4i_ptr)(lds_dst),
      0, 0);
#else
  unsigned lds_off = (unsigned)(unsigned long long)
      (__attribute__((address_space(3))) char*)lds_dst;
  unsigned long long ga = (unsigned long long)gsrc;
  asm volatile("global_load_async_to_lds_b128 %0, %1, off"
               :: "v"(lds_off), "v"(ga) : "memory");
#endif
}

__device__ __forceinline__ void wait_async_zero() {
#if __has_builtin(__builtin_amdgcn_s_wait_asynccnt)
  __builtin_amdgcn_s_wait_asynccnt(0);
#else
  asm volatile("s_wait_asynccnt 0" ::: "memory");
#endif
}

// ---- fast tanh: gfx1250 v_tanh_f32 (builtin preferred, asm fallback) -------
__device__ __forceinline__ float fast_tanh(float x) {
#if __has_builtin(__builtin_amdgcn_tanhf)
  return __builtin_amdgcn_tanhf(x);
#elif __has_builtin(__builtin_amdgcn_tanh_f32)
  return __builtin_amdgcn_tanh_f32(x);
#else
  float r;
  asm("v_tanh_f32 %0, %1" : "=v"(r) : "v"(x));
  return r;
#endif
}

// ---------------------------------------------------------------------------
__global__ __launch_bounds__(kThreads)
void softrank_kernel(const float* __restrict__ x, float* __restrict__ out) {
  __shared__ float smem[kN * kC];          // 64 KB: one whole batch

  const int b    = blockIdx.x / kJBlks;
  const int jblk = blockIdx.x % kJBlks;
  const int tid  = threadIdx.x;

  const float* xb = x + (size_t)b * kN * kC;

  // ---- stage batch into LDS: 4096 x 16B chunks, 16 per thread -------------
  constexpr int kChunks = (kN * kC) / 4;   // float4 chunks
#pragma unroll
  for (int k = 0; k < kChunks / kThreads; ++k) {
    const int idx4 = tid + k * kThreads;
    async_cp16(&smem[idx4 * 4], xb + idx4 * 4);
  }
  wait_async_zero();
  __syncthreads();

  // ---- each thread: one j row, 2 channels ---------------------------------
  const int cg = tid & (kCG - 1);          // 0..7
  const int jl = tid >> 3;                 // 0..31
  const int j  = jblk * kJPerBlk + jl;
  const int c0 = cg * kCPT;

  const float cj0 = kAlphaH * smem[j * kC + c0];
  const float cj1 = kAlphaH * smem[j * kC + c0 + 1];

  float acc0 = 0.0f, acc1 = 0.0f;

#pragma unroll 8
  for (int i = 0; i < kN; ++i) {
    const float2 v = *reinterpret_cast<const float2*>(&smem[i * kC + c0]);
    const float z0 = __builtin_fmaf(v.x, -kAlphaH, cj0);
    const float z1 = __builtin_fmaf(v.y, -kAlphaH, cj1);
    acc0 += fast_tanh(z0);                 // v_tanh_f32 (trans pipe)
    acc1 += fast_tanh(z1);
  }

  // out = 0.5 + (1/(2N)) * sum(tanh)
  float2 o;
  o.x = __builtin_fmaf(acc0, 0.5f / (float)kN, 0.5f);
  o.y = __builtin_fmaf(acc1, 0.5f / (float)kN, 0.5f);

  *reinterpret_cast<float2*>(&out[((size_t)b * kN + j) * kC + c0]) = o;
}

// ---------------------------------------------------------------------------
extern "C" void kernel_launch(void* const* d_in, const int* in_sizes, int n_in,
                              void* d_out, int out_size, void* d_ws, size_t ws_size,
                              hipStream_t stream) {
  (void)in_sizes; (void)n_in; (void)out_size; (void)d_ws; (void)ws_size;
  const float* x = (const float*)d_in[0];
  float* out = (float*)d_out;
  softrank_kernel<<<dim3(kB * kJBlks), dim3(kThreads), 0, stream>>>(x, out);
}